// MCLLoss_2808908612055
// MI455X (gfx1250) — compile-verified
//
#include <hip/hip_runtime.h>

typedef __attribute__((ext_vector_type(2))) float v2f;
typedef __attribute__((ext_vector_type(8))) float v8f;

// Per-element classification/accumulation.
// Ranges: 0:[-1,1] 1:[-1,-0.5] 2:[-0.5,0] 3:[0,0.5] 4:[0.5,1]
// Sums/counts are per-mask (overlapping); class id = highest matching range.
__device__ __forceinline__ void mcl_elem(float r, float d,
                                         float* __restrict__ s,
                                         float* __restrict__ c,
                                         float* __restrict__ mout) {
  bool in0 = (r >= -1.0f) & (r <= 1.0f);
  bool in1 = (r >= -1.0f) & (r <= -0.5f);
  bool in2 = (r >= -0.5f) & (r <= 0.0f);
  bool in3 = (r >= 0.0f) & (r <= 0.5f);
  bool in4 = (r >= 0.5f) & (r <= 1.0f);
  if (in0) { s[0] += d; c[0] += 1.0f; }
  if (in1) { s[1] += d; c[1] += 1.0f; }
  if (in2) { s[2] += d; c[2] += 1.0f; }
  if (in3) { s[3] += d; c[3] += 1.0f; }
  if (in4) { s[4] += d; c[4] += 1.0f; }
  int j = in4 ? 4 : (in3 ? 3 : (in2 ? 2 : (in1 ? 1 : 0)));
  *mout = 0.5f * (float)j - 1.0f;  // 2*j/(5-1) - 1
}

__global__ void mcl_main(const float* __restrict__ pre,
                         const float* __restrict__ real,
                         float* __restrict__ mask_out,
                         float* __restrict__ ws,
                         long long n) {
  float s[5] = {0.f, 0.f, 0.f, 0.f, 0.f};
  float c[5] = {0.f, 0.f, 0.f, 0.f, 0.f};

  const long long n4 = n >> 2;
  const long long tid = (long long)blockIdx.x * blockDim.x + threadIdx.x;
  const long long stride = (long long)gridDim.x * blockDim.x;

  const float4* __restrict__ p4 = (const float4*)pre;
  const float4* __restrict__ r4 = (const float4*)real;
  float4* __restrict__ m4 = (float4*)mask_out;

  // Vectorized streaming: global_load_b128 x2, global_store_b128 x1 per iter.
  for (long long i = tid; i < n4; i += stride) {
    float4 pv = p4[i];
    float4 rv = r4[i];
    float4 mo;
    mcl_elem(rv.x, fabsf(pv.x - rv.x), s, c, &mo.x);
    mcl_elem(rv.y, fabsf(pv.y - rv.y), s, c, &mo.y);
    mcl_elem(rv.z, fabsf(pv.z - rv.z), s, c, &mo.z);
    mcl_elem(rv.w, fabsf(pv.w - rv.w), s, c, &mo.w);
    m4[i] = mo;
  }
  // Scalar tail (n not multiple of 4).
  for (long long i = (n4 << 2) + tid; i < n; i += stride) {
    float r = real[i];
    float d = fabsf(pre[i] - r);
    float mo;
    mcl_elem(r, d, s, c, &mo);
    mask_out[i] = mo;
  }

  // Wave32 butterfly: every lane ends up with the full wave totals.
  #pragma unroll
  for (int off = 16; off > 0; off >>= 1) {
    #pragma unroll
    for (int j = 0; j < 5; ++j) {
      s[j] += __shfl_xor(s[j], off, 32);
      c[j] += __shfl_xor(c[j], off, 32);
    }
  }

  // 8 wave-partials (16 columns each: 5 sums, 5 counts, 6 zero pad) in LDS.
  __shared__ float P[8][16];
  const int lane = threadIdx.x & 31;
  const int wv = threadIdx.x >> 5;
  if (lane == 0) {
    #pragma unroll
    for (int j = 0; j < 5; ++j) { P[wv][j] = s[j]; P[wv][5 + j] = c[j]; }
    #pragma unroll
    for (int j = 10; j < 16; ++j) P[wv][j] = 0.0f;
  }
  __syncthreads();

  if (wv == 0) {
    // Block reduction via V_WMMA_F32_16X16X4_F32 with A == ones:
    // D[m][n] = sum_k B[k][n], independent of B's physical K permutation.
    // Two chained WMMAs sum all 8 wave-partials per column n.
    const int col = lane & 15;
    const bool hi = lane >= 16;
    v2f a; a.x = 1.0f; a.y = 1.0f;
    v2f b0, b1;
    b0.x = hi ? P[2][col] : P[0][col];
    b0.y = hi ? P[3][col] : P[1][col];
    b1.x = hi ? P[6][col] : P[4][col];
    b1.y = hi ? P[7][col] : P[5][col];
    v8f acc = {};
    acc = __builtin_amdgcn_wmma_f32_16x16x4_f32(false, a, false, b0,
                                                (short)0, acc, false, false);
    acc = __builtin_amdgcn_wmma_f32_16x16x4_f32(false, a, false, b1,
                                                (short)0, acc, false, false);
    // D VGPR0, lanes 0..15 -> row M=0, N=lane. Columns 0..9 are our totals.
    if (lane < 10) atomicAdd(&ws[lane], acc[0]);
  }
}

__global__ void mcl_finalize(const float* __restrict__ ws,
                             float* __restrict__ out) {
  const int t = threadIdx.x;
  __shared__ float li[5];
  if (t < 5) {
    float sum = ws[t];
    float cnt = ws[5 + t];
    float l = (cnt > 0.0f) ? (sum / cnt) * 0.2f : 0.0f;
    li[t] = l;
    out[1 + t] = l;
  }
  __syncthreads();
  if (t == 0) out[0] = (((li[0] + li[1]) + li[2]) + li[3]) + li[4];
}

extern "C" void kernel_launch(void* const* d_in, const int* in_sizes, int n_in,
                              void* d_out, int out_size, void* d_ws, size_t ws_size,
                              hipStream_t stream) {
  const float* pre = (const float*)d_in[0];
  const float* real = (const float*)d_in[1];
  float* out = (float*)d_out;
  float* ws = (float*)d_ws;
  const long long n = (long long)in_sizes[0];

  // ws[0..4] = masked diff sums, ws[5..9] = counts (as float; exact <= 2^24).
  hipMemsetAsync(d_ws, 0, 10 * sizeof(float), stream);

  const int threads = 256;  // 8 wave32s per block
  long long n4 = n >> 2;
  long long want = (n4 + threads - 1) / threads;
  int blocks = (int)(want < 1 ? 1 : (want > 8192 ? 8192 : want));

  mcl_main<<<blocks, threads, 0, stream>>>(pre, real, out + 6, ws, n);
  mcl_finalize<<<1, 32, 0, stream>>>(ws, out);
}